// ColliderModel_38087769980972
// MI455X (gfx1250) — compile-verified
//
#include <hip/hip_runtime.h>

// Collider binary-tree sampler (DEPTH=9, 510 vars/row, level-major).
// One wave32 per row. CDNA5 path:
//   - global_load_async_to_lds_b64: stage the 2040B row into a per-wave LDS tile
//     (ASYNCcnt, 8B-aligned chunks, imm offset applies to LDS and global sides)
//   - global_store_async_from_lds_b64: level-0 output == level-0 eps, so the
//     first 1024B of each output row is a pure LDS->global async memcpy
//   - levels 1-7: intra-lane pair sums + wave32 ds_bpermute for widths<=16,
//     nontemporal b64 stores (single-pass stream, keep 192MB L2 clean)
// Memory-bound: ~1.07 GB traffic -> ~46 us roofline at 23.3 TB/s.

typedef float v2f __attribute__((ext_vector_type(2)));

#define SIG 0.1f
#define TV  510

__global__ __launch_bounds__(256) void collider_tree_async_kernel(
    const float* __restrict__ eps, float* __restrict__ out, int n)
{
    __shared__ float tile[8][512];          // 2048 B staging buffer per wave

    const int gtid = blockIdx.x * 256 + threadIdx.x;
    const int row  = gtid >> 5;
    const int lane = gtid & 31;
    const int w    = threadIdx.x >> 5;
    if (row >= n) return;

    const float* e = eps + (size_t)row * TV;   // row stride 2040 B (8B aligned)
    float*       o = out + (size_t)row * TV;
    float*       t = &tile[w][0];

    // ---- async stage: 2040 B row -> LDS as 255 aligned b64 chunks ----------
    // chunk c = lane + 32k lives at byte offset c*8 on both sides; the 24-bit
    // instruction offset is added to BOTH the LDS and the global address.
    unsigned long long ga   = (unsigned long long)(uintptr_t)e + (unsigned)(lane * 8);
    unsigned int       ldsb = (unsigned int)(uintptr_t)t       + (unsigned)(lane * 8);

    asm volatile("global_load_async_to_lds_b64 %0, %1, off"             :: "v"(ldsb), "v"(ga) : "memory");
    asm volatile("global_load_async_to_lds_b64 %0, %1, off offset:256"  :: "v"(ldsb), "v"(ga) : "memory");
    asm volatile("global_load_async_to_lds_b64 %0, %1, off offset:512"  :: "v"(ldsb), "v"(ga) : "memory");
    asm volatile("global_load_async_to_lds_b64 %0, %1, off offset:768"  :: "v"(ldsb), "v"(ga) : "memory");
    asm volatile("global_load_async_to_lds_b64 %0, %1, off offset:1024" :: "v"(ldsb), "v"(ga) : "memory");
    asm volatile("global_load_async_to_lds_b64 %0, %1, off offset:1280" :: "v"(ldsb), "v"(ga) : "memory");
    asm volatile("global_load_async_to_lds_b64 %0, %1, off offset:1536" :: "v"(ldsb), "v"(ga) : "memory");
    if (lane < 31)  // chunk 255 would touch byte 2040..2047: past the row
        asm volatile("global_load_async_to_lds_b64 %0, %1, off offset:1792" :: "v"(ldsb), "v"(ga) : "memory");

    asm volatile("s_wait_asynccnt 0" ::: "memory");

    // ---- level 0 (width 256): x0 == eps0, async memcpy LDS -> global -------
    unsigned long long go = (unsigned long long)(uintptr_t)o + (unsigned)(lane * 8);
    asm volatile("global_store_async_from_lds_b64 %0, %1, off"            :: "v"(go), "v"(ldsb) : "memory");
    asm volatile("global_store_async_from_lds_b64 %0, %1, off offset:256" :: "v"(go), "v"(ldsb) : "memory");
    asm volatile("global_store_async_from_lds_b64 %0, %1, off offset:512" :: "v"(go), "v"(ldsb) : "memory");
    asm volatile("global_store_async_from_lds_b64 %0, %1, off offset:768" :: "v"(go), "v"(ldsb) : "memory");
    // s_endpgm performs an implicit wait-idle, so these complete before exit.

    // ---- level 1 (width 128): lane holds cols 4L..4L+3 ---------------------
    v2f a[4];
#pragma unroll
    for (int j = 0; j < 4; ++j) a[j] = *(const v2f*)(t + lane * 8 + 2 * j);

    v2f x1[2];
#pragma unroll
    for (int j = 0; j < 2; ++j) {
        v2f ev = *(const v2f*)(t + 256 + lane * 4 + 2 * j);
        v2f m;
        m.x = a[2 * j].x     + a[2 * j].y;      // parents 8L+4j,   8L+4j+1
        m.y = a[2 * j + 1].x + a[2 * j + 1].y;  // parents 8L+4j+2, 8L+4j+3
        x1[j] = m + SIG * ev;
        __builtin_nontemporal_store(x1[j], (v2f*)(o + 256) + lane * 2 + j);
    }

    // ---- level 2 (width 64): lane holds cols 2L, 2L+1 ----------------------
    v2f ev2 = *(const v2f*)(t + 384 + lane * 2);
    v2f x2;
    x2.x = (x1[0].x + x1[0].y) + SIG * ev2.x;
    x2.y = (x1[1].x + x1[1].y) + SIG * ev2.y;
    __builtin_nontemporal_store(x2, (v2f*)(o + 384) + lane);

    // ---- level 3 (width 32): lane holds col L ------------------------------
    float x3 = (x2.x + x2.y) + SIG * t[448 + lane];
    __builtin_nontemporal_store(x3, o + 448 + lane);

    // ---- level 4 (width 16): cross-lane via wave32 shuffle -----------------
    float p0 = __shfl(x3, (lane * 2) & 31, 32);
    float p1 = __shfl(x3, (lane * 2 + 1) & 31, 32);
    float x4 = 0.0f;
    if (lane < 16) {
        x4 = (p0 + p1) + SIG * t[480 + lane];
        __builtin_nontemporal_store(x4, o + 480 + lane);
    }

    // ---- level 5 (width 8) -------------------------------------------------
    float q0 = __shfl(x4, (lane * 2) & 31, 32);
    float q1 = __shfl(x4, (lane * 2 + 1) & 31, 32);
    float x5 = 0.0f;
    if (lane < 8) {
        x5 = (q0 + q1) + SIG * t[496 + lane];
        __builtin_nontemporal_store(x5, o + 496 + lane);
    }

    // ---- level 6 (width 4) -------------------------------------------------
    float r0 = __shfl(x5, (lane * 2) & 31, 32);
    float r1 = __shfl(x5, (lane * 2 + 1) & 31, 32);
    float x6 = 0.0f;
    if (lane < 4) {
        x6 = (r0 + r1) + SIG * t[504 + lane];
        __builtin_nontemporal_store(x6, o + 504 + lane);
    }

    // ---- level 7 (width 2) -------------------------------------------------
    float s0 = __shfl(x6, (lane * 2) & 31, 32);
    float s1 = __shfl(x6, (lane * 2 + 1) & 31, 32);
    if (lane < 2) {
        float x7 = (s0 + s1) + SIG * t[508 + lane];
        __builtin_nontemporal_store(x7, o + 508 + lane);
    }
}

extern "C" void kernel_launch(void* const* d_in, const int* in_sizes, int n_in,
                              void* d_out, int out_size, void* d_ws, size_t ws_size,
                              hipStream_t stream) {
    const float* eps = (const float*)d_in[0];
    float* out = (float*)d_out;
    const int n = in_sizes[0] / TV;            // rows

    const int rows_per_block = 256 / 32;       // one wave32 per row
    const int blocks = (n + rows_per_block - 1) / rows_per_block;
    collider_tree_async_kernel<<<blocks, 256, 0, stream>>>(eps, out, n);
}